// AELoss_29094108463534
// MI455X (gfx1250) — compile-verified
//
#include <hip/hip_runtime.h>
#include <hip/hip_bf16.h>
#include <math.h>

// Problem constants (match reference)
#define NPARTS 17
#define EDIM   32
#define NPEOPLE 20
#define NBATCH 16
#define HH 128
#define WW 128
#define PK   (NPEOPLE * NPARTS)   // 340 stacked (person,part) rows
#define NVE  (PK * EDIM)          // 10880 gathered floats per batch
#define HW   (HH * WW)            // 16384

typedef float v2f __attribute__((ext_vector_type(2)));
typedef float v8f __attribute__((ext_vector_type(8)));

// One workgroup (1024 threads = 32 wave32s) per batch image.
__global__ __launch_bounds__(1024)
void aeloss_kernel(const float* __restrict__ ebd,   // [B, 17*32, H, W]
                   const float* __restrict__ kpts,  // [B, P, 17, 2]
                   float* __restrict__ out) {       // [1]
    __shared__ float vecs[PK * EDIM];      // V: row (p*17+k), col d   (43.5 KB)
    __shared__ float means[NPEOPLE * EDIM];
    __shared__ int   lin[PK];              // y*W + x per (p,k)
    __shared__ float red_pull[32];
    __shared__ float red_push[32];

    const int b    = blockIdx.x;
    const int tid  = threadIdx.x;
    const int wave = tid >> 5;
    const int lane = tid & 31;
    const long ebd_base = (long)b * (NPARTS * EDIM) * HW;

    // ---- 1) spatial linear index per (person, part) -------------------------
    for (int i = tid; i < PK; i += blockDim.x) {
        int p = i / NPARTS, k = i % NPARTS;
        const float* kp = kpts + (((long)b * NPEOPLE + p) * NPARTS + k) * 2;
        int y = (int)floorf(kp[0] * (float)HH);
        int x = (int)floorf(kp[1] * (float)WW);
        y = y < 0 ? 0 : (y > HH - 1 ? HH - 1 : y);
        x = x < 0 ? 0 : (x > WW - 1 ? WW - 1 : x);
        lin[i] = y * WW + x;
    }
    __syncthreads();

    // ---- 2) scattered gather: each element is its own cacheline; spread
    //         10880 independent loads across 32 waves for MLP ----------------
    for (int i = tid; i < NVE; i += blockDim.x) {
        int pk = i / EDIM, d = i % EDIM;
        int k  = pk % NPARTS;
        vecs[i] = ebd[ebd_base + (long)(k * EDIM + d) * HW + lin[pk]];
    }
    __syncthreads();

    // ---- 3) means = S(20x340) x V(340x32) via V_WMMA_F32_16X16X4_F32 -------
    // 4 waves, one 16x16 output tile each (p-tiles {0,16} x d-tiles {0,16}).
    // A (16x4 f32) layout: lanes0-15 M=0..15 hold K=0(K=1 in v1); lanes16-31
    // hold K=2/3.  D layout: VGPR r -> M=r (lanes0-15) / M=r+8 (lanes16-31).
    if (wave < 4) {
        const int p_base = (wave & 1) * 16;
        const int d_base = (wave >> 1) * 16;
        const int half   = lane >> 4;    // 0: lanes 0-15, 1: lanes 16-31
        const int l      = lane & 15;
        const int prow   = p_base + l;   // A-matrix row for this lane
        v8f c = {0.f, 0.f, 0.f, 0.f, 0.f, 0.f, 0.f, 0.f};
        for (int j = 0; j < PK / 4; ++j) {
            int k0 = 4 * j + 2 * half;   // this lane's two K columns: k0, k0+1
            v2f a, bm;
            // indicator/17: column p'*17+k belongs to person p' = col/17
            a.x = ((k0    ) / NPARTS == prow) ? (1.0f / (float)NPARTS) : 0.0f;
            a.y = ((k0 + 1) / NPARTS == prow) ? (1.0f / (float)NPARTS) : 0.0f;
            // B (4x16): K-rows striped like A, N = l
            bm.x = vecs[(k0    ) * EDIM + d_base + l];
            bm.y = vecs[(k0 + 1) * EDIM + d_base + l];
            c = __builtin_amdgcn_wmma_f32_16x16x4_f32(
                    false, a, false, bm, (short)0, c, false, false);
        }
        #pragma unroll
        for (int r = 0; r < 8; ++r) {
            int m = p_base + r + 8 * half;
            if (m < NPEOPLE) means[m * EDIM + d_base + l] = c[r];
        }
    }
    __syncthreads();

    // ---- 4) pull: sum |v - mean| over (p,k,d);  push: sum |m_p - m_q| ------
    float lp = 0.f;
    for (int i = tid; i < NVE; i += blockDim.x) {
        int pk = i / EDIM, d = i % EDIM;
        int p  = pk / NPARTS;
        lp += fabsf(vecs[i] - means[p * EDIM + d]);
    }
    float lq = 0.f;
    for (int i = tid; i < NPEOPLE * NPEOPLE * EDIM; i += blockDim.x) {
        int d  = i % EDIM;
        int pq = i / EDIM;
        int p  = pq / NPEOPLE, q = pq % NPEOPLE;
        lq += fabsf(means[p * EDIM + d] - means[q * EDIM + d]);
    }

    // wave-level tree reduce (wave32)
    #pragma unroll
    for (int off = 16; off > 0; off >>= 1) {
        lp += __shfl_down(lp, off, 32);
        lq += __shfl_down(lq, off, 32);
    }
    if (lane == 0) { red_pull[wave] = lp; red_push[wave] = lq; }
    __syncthreads();

    if (wave == 0) {
        const int nwaves = blockDim.x >> 5;   // 32
        lp = (lane < nwaves) ? red_pull[lane] : 0.f;
        lq = (lane < nwaves) ? red_push[lane] : 0.f;
        #pragma unroll
        for (int off = 16; off > 0; off >>= 1) {
            lp += __shfl_down(lp, off, 32);
            lq += __shfl_down(lq, off, 32);
        }
        if (lane == 0) {
            float pull_b = lp / (float)(NPARTS * EDIM);           // sum_p mean_{k,d}
            float push_b = lq / (float)(NPEOPLE * NPEOPLE * EDIM);// mean_{p,q,d}
            // final = (mean_b push + mean_b pull) / 2
            atomicAdd(out, (pull_b + push_b) * (1.0f / (2.0f * (float)NBATCH)));
        }
    }
}

extern "C" void kernel_launch(void* const* d_in, const int* in_sizes, int n_in,
                              void* d_out, int out_size, void* d_ws, size_t ws_size,
                              hipStream_t stream) {
    const float* ebd  = (const float*)d_in[0];  // [16, 544, 128, 128] f32
    const float* kpts = (const float*)d_in[1];  // [16, 20, 17, 2] f32
    float* out = (float*)d_out;                 // scalar f32
    (void)in_sizes; (void)n_in; (void)out_size; (void)d_ws; (void)ws_size;

    hipMemsetAsync(out, 0, sizeof(float), stream);  // capture-safe memset node
    aeloss_kernel<<<NBATCH, 1024, 0, stream>>>(ebd, kpts, out);
}